// BarrelShifterRight48_23407571764135
// MI455X (gfx1250) — compile-verified
//
#include <hip/hip_runtime.h>

typedef unsigned int u32;
typedef unsigned long long u64;
typedef __attribute__((ext_vector_type(4))) unsigned int v4u;
typedef __attribute__((ext_vector_type(8))) int v8i;
typedef __attribute__((ext_vector_type(4))) int v4i;

#define ROWS_PER_BLOCK 256
#define XPAD 52   // LDS row stride for X, in floats (48 data + 4 pad -> 208B, 16B aligned, 2-way banks)
#define SPAD 8    // LDS row stride for shift, in floats

#if defined(__has_builtin)
#  if __has_builtin(__builtin_amdgcn_tensor_load_to_lds)
#    define HAVE_TDM 1
#  endif
#endif
#ifndef HAVE_TDM
#  define HAVE_TDM 0
#endif

#if HAVE_TDM
// Issue one TDM descriptor: copy `nrows` rows of `row_dwords` dwords each from
// a contiguous global region to LDS, advancing LDS by lds_stride_dw dwords per
// row (iterate mode -> inserts LDS padding; global stays perfectly linear).
__device__ __forceinline__ void tdm_load_rows(u32 lds_off, u64 gaddr,
                                              u32 row_dwords, u32 lds_stride_dw,
                                              u32 nrows) {
  // ---- D# group 0 (ISA 8.3): count=1, lds_addr, global_addr[56:0], type=2
  v4u g0;
  g0.x = 1u;                                            // count=1, user mode
  g0.y = lds_off;                                       // LDS byte address
  g0.z = (u32)(gaddr & 0xFFFFFFFFull);                  // global_addr lo32
  g0.w = (u32)((gaddr >> 32) & 0x01FFFFFFull) | (2u << 30); // ga[56:32] | type=2

  // ---- D# group 1 (ISA 8.4): data_size=4B(code 2), iterate_enable=1,
  // tensor_dim0=row_dwords, tensor_dim1=1, tile_dim0=row_dwords, tile_dim1=1,
  // tensor_dim0_stride=row_dwords. workgroup_mask=0 (not in a cluster).
  v8i g1;
  g1[0] = (int)((2u << 16) | (1u << 19));               // data_size | iterate_enable
  g1[1] = (int)((row_dwords & 0xFFFFu) << 16);          // tensor_dim0 lo16 in [63:48]
  g1[2] = (int)((row_dwords >> 16) | (1u << 16));       // tensor_dim0 hi16 | tensor_dim1=1
  g1[3] = (int)((row_dwords & 0xFFFFu) << 16);          // tensor_dim1 hi=0 | tile_dim0
  g1[4] = 1;                                            // tile_dim1=1, tile_dim2=0
  g1[5] = (int)row_dwords;                              // tensor_dim0_stride lo32
  g1[6] = 0;                                            // stride hi16, dim1_stride lo16
  g1[7] = 0;

  // ---- D# group 2 (ISA 8.5, iterate mode): lds/global increments + count
  v4i g2;
  g2[0] = 0;                                            // tensor_dim2 (unused)
  g2[1] = (int)lds_stride_dw;                           // lds_addr_increment (data_size units)
  g2[2] = (int)row_dwords;                              // global_addr_increment lo32
  g2[3] = (int)((nrows - 1u) << 16);                    // ga_inc hi16=0 | iterate_count
  v4i g3 = {0, 0, 0, 0};

  // This toolchain exposes the 6-arg builtin: one extra (reserved) SGPR group.
  v8i g4 = {0, 0, 0, 0, 0, 0, 0, 0};

  __builtin_amdgcn_tensor_load_to_lds(g0, g1, g2, g3, g4, /*cpol=*/0);
}
#endif

__global__ __launch_bounds__(ROWS_PER_BLOCK)
void barrel48_kernel(const float* __restrict__ X, const float* __restrict__ S,
                     float* __restrict__ out, int B) {
  const int t = threadIdx.x;
  const int tileBase = (int)blockIdx.x * ROWS_PER_BLOCK;
  const long long row = (long long)tileBase + t;

#if HAVE_TDM
  __shared__ float ldsX[ROWS_PER_BLOCK * XPAD];  // 53,248 B
  __shared__ float ldsS[ROWS_PER_BLOCK * SPAD];  //  8,192 B

  int rem = B - tileBase;
  u32 nrows = (u32)(rem < ROWS_PER_BLOCK ? rem : ROWS_PER_BLOCK);

  // Wave-uniform branch (readfirstlane -> scalar compare) so waves 1..7 take a
  // scalar skip and never issue the tensor ops (TENSOR ops ignore EXEC).
  if (__builtin_amdgcn_readfirstlane((unsigned)t >> 5) == 0u) {
    tdm_load_rows((u32)(size_t)(void*)ldsX,
                  (u64)(size_t)(const void*)(X + (size_t)tileBase * 48),
                  48u, XPAD, nrows);
    tdm_load_rows((u32)(size_t)(void*)ldsS,
                  (u64)(size_t)(const void*)(S + (size_t)tileBase * 6),
                  6u, SPAD, nrows);
    __builtin_amdgcn_s_wait_tensorcnt(0);       // both TDM ops complete
  }
  __syncthreads();
#endif

  if (row >= B) return;

  // ---- pack the 48 float bits into a uint64 (X[0] -> bit 47)
  u64 bits = 0;
#if HAVE_TDM
  const float4* xr = reinterpret_cast<const float4*>(&ldsX[(size_t)t * XPAD]);
#else
  const float4* xr = reinterpret_cast<const float4*>(X + (size_t)row * 48);
#endif
#pragma unroll
  for (int k = 0; k < 12; ++k) {
    float4 v = xr[k];
    u64 nib = ((u64)(v.x != 0.0f) << 3) | ((u64)(v.y != 0.0f) << 2) |
              ((u64)(v.z != 0.0f) << 1) |  (u64)(v.w != 0.0f);
    bits = (bits << 4) | nib;
  }

  // ---- shift amount: layer 0 has weight 32 ... layer 5 weight 1
#if HAVE_TDM
  const float* sr = &ldsS[(size_t)t * SPAD];
#else
  const float* sr = S + (size_t)row * 6;
#endif
  u32 s = 0;
#pragma unroll
  for (int l = 0; l < 6; ++l) s |= (u32)(sr[l] != 0.0f) << (5 - l);

  // ---- barrel shift + sticky (OR of all shifted-out bits = low s bits)
  u64 res   = bits >> s;                        // s in [0,63]
  u64 mask  = ((u64)1 << s) - 1ull;             // s==0 -> 0
  float sticky = (bits & mask) ? 1.0f : 0.0f;

  // ---- write 48 result bits (12 x b128) + sticky
  float* orow = out + (size_t)row * 48;
#pragma unroll
  for (int k = 0; k < 12; ++k) {
    int hi = 47 - 4 * k;
    float4 o;
    o.x = (float)((res >> hi)       & 1ull);
    o.y = (float)((res >> (hi - 1)) & 1ull);
    o.z = (float)((res >> (hi - 2)) & 1ull);
    o.w = (float)((res >> (hi - 3)) & 1ull);
    reinterpret_cast<float4*>(orow)[k] = o;
  }
  out[(size_t)B * 48 + (size_t)row] = sticky;
}

extern "C" void kernel_launch(void* const* d_in, const int* in_sizes, int n_in,
                              void* d_out, int out_size, void* d_ws, size_t ws_size,
                              hipStream_t stream) {
  const float* X = (const float*)d_in[0];   // [B,48] float32 bits
  const float* S = (const float*)d_in[1];   // [B,6]  float32 shift bits
  float* out = (float*)d_out;               // [B*48] current ++ [B] sticky
  const int B = in_sizes[0] / 48;
  const int blocks = (B + ROWS_PER_BLOCK - 1) / ROWS_PER_BLOCK;
  hipLaunchKernelGGL(barrel48_kernel, dim3(blocks), dim3(ROWS_PER_BLOCK), 0, stream,
                     X, S, out, B);
}